// QuantumNATGen200_65481071400427
// MI455X (gfx1250) — compile-verified
//
#include <hip/hip_runtime.h>
#include <hip/hip_bf16.h>
#include <math.h>

// ---------------------------------------------------------------------------
// QuantumNATGen200 pipeline for gfx1250 (MI455X).
// Attention (the only O(B^2) stage) uses V_WMMA_F32_16X16X4_F32 in a
// two-phase flash loop: phase 1 = per-lane running row-max (no shuffles in
// the hot loop), phase 2 = exp + O += P·g. Cross-lane reductions happen
// exactly twice per row-tile instead of per chunk, keeping the DS pipe idle.
// ---------------------------------------------------------------------------

typedef __attribute__((ext_vector_type(2))) float v2f;
typedef __attribute__((ext_vector_type(8))) float v8f;

#define BATCH 4096

// workspace layout (float offsets); total ~12.99M floats (~52 MB)
#define WS_H1     0           // (B, 8,16,16)  = 8388608
#define WS_H2     8388608     // (B,16, 8, 8)  = 4194304
#define WS_FEATS  12582912    // (B,32)        = 131072
#define WS_G      12713984    // (B,16)        = 65536
#define WS_Q      12779520    // (B,16)        = 65536
#define WS_K      12845056    // (B,16)        = 65536
#define WS_ATTN   12910592    // (B,16)        = 65536
#define WS_PRE    12976128    // (B,2)         = 8192
#define WS_ALPHA  12984320    // 1

// ---------------- conv1 (1->8, 3x3 SAME, relu, maxpool2) -------------------
__global__ __launch_bounds__(256) void conv1_kernel(
    const float* __restrict__ x, const float* __restrict__ w,
    const float* __restrict__ b, float* __restrict__ h1)
{
  __shared__ float in_s[1024];   // 1x32x32
  __shared__ float w_s[72];      // 8x1x3x3
  __shared__ float b_s[8];
  const int img = blockIdx.x;
  const int t = threadIdx.x;
  for (int i = t; i < 1024; i += 256) in_s[i] = x[img * 1024 + i];
  if (t < 72) w_s[t] = w[t];
  if (t < 8)  b_s[t] = b[t];
  __syncthreads();
  for (int p = t; p < 2048; p += 256) {            // 8 oc * 16 * 16 pooled
    const int oc = p >> 8, pos = p & 255;
    const int oy = pos >> 4, ox = pos & 15;
    float mx = -3.0e38f;
#pragma unroll
    for (int dy = 0; dy < 2; ++dy)
#pragma unroll
      for (int dx = 0; dx < 2; ++dx) {
        const int cy = oy * 2 + dy, cx = ox * 2 + dx;
        float acc = b_s[oc];
#pragma unroll
        for (int ky = 0; ky < 3; ++ky) {
          const int iy = cy + ky - 1;
          if (iy < 0 || iy > 31) continue;
#pragma unroll
          for (int kx = 0; kx < 3; ++kx) {
            const int ix = cx + kx - 1;
            if (ix < 0 || ix > 31) continue;
            acc += in_s[iy * 32 + ix] * w_s[oc * 9 + ky * 3 + kx];
          }
        }
        acc = fmaxf(acc, 0.0f);
        mx = fmaxf(mx, acc);
      }
    h1[img * 2048 + p] = mx;                       // (img, oc, y, x)
  }
}

// ---------------- conv2 (8->16, 3x3 SAME, relu, maxpool2) ------------------
__global__ __launch_bounds__(256) void conv2_kernel(
    const float* __restrict__ h1, const float* __restrict__ w,
    const float* __restrict__ b, float* __restrict__ h2)
{
  __shared__ float in_s[2048];   // 8x16x16
  __shared__ float w_s[1152];    // 16x8x3x3
  __shared__ float b_s[16];
  const int img = blockIdx.x;
  const int t = threadIdx.x;
  for (int i = t; i < 2048; i += 256) in_s[i] = h1[img * 2048 + i];
  for (int i = t; i < 1152; i += 256) w_s[i] = w[i];
  if (t < 16) b_s[t] = b[t];
  __syncthreads();
  for (int p = t; p < 1024; p += 256) {            // 16 oc * 8 * 8 pooled
    const int oc = p >> 6, pos = p & 63;
    const int oy = pos >> 3, ox = pos & 7;
    float mx = -3.0e38f;
#pragma unroll
    for (int dy = 0; dy < 2; ++dy)
#pragma unroll
      for (int dx = 0; dx < 2; ++dx) {
        const int cy = oy * 2 + dy, cx = ox * 2 + dx;
        float acc = b_s[oc];
        for (int ic = 0; ic < 8; ++ic) {
#pragma unroll
          for (int ky = 0; ky < 3; ++ky) {
            const int iy = cy + ky - 1;
            if (iy < 0 || iy > 15) continue;
#pragma unroll
            for (int kx = 0; kx < 3; ++kx) {
              const int ix = cx + kx - 1;
              if (ix < 0 || ix > 15) continue;
              acc += in_s[ic * 256 + iy * 16 + ix] *
                     w_s[oc * 72 + ic * 9 + ky * 3 + kx];
            }
          }
        }
        acc = fmaxf(acc, 0.0f);
        mx = fmaxf(mx, acc);
      }
    h2[img * 1024 + p] = mx;                       // (img, oc, y, x)
  }
}

// ------------- conv3 (16->32, 3x3 SAME, relu) + spatial mean ---------------
__global__ __launch_bounds__(256) void conv3_kernel(
    const float* __restrict__ h2, const float* __restrict__ w,
    const float* __restrict__ b, float* __restrict__ feats)
{
  __shared__ float in_s[1024];   // 16x8x8
  __shared__ float w_s[4608];    // 32x16x3x3
  __shared__ float b_s[32];
  const int img = blockIdx.x;
  const int t = threadIdx.x;
  for (int i = t; i < 1024; i += 256) in_s[i] = h2[img * 1024 + i];
  for (int i = t; i < 4608; i += 256) w_s[i] = w[i];
  if (t < 32) b_s[t] = b[t];
  __syncthreads();
  const int oc = t >> 3;         // 0..31
  const int pg = t & 7;          // column handled by this thread
  float sum = 0.0f;
  for (int y = 0; y < 8; ++y) {  // pixel (y, pg)
    float acc = b_s[oc];
    for (int ic = 0; ic < 16; ++ic) {
#pragma unroll
      for (int ky = 0; ky < 3; ++ky) {
        const int iy = y + ky - 1;
        if (iy < 0 || iy > 7) continue;
#pragma unroll
        for (int kx = 0; kx < 3; ++kx) {
          const int ix = pg + kx - 1;
          if (ix < 0 || ix > 7) continue;
          acc += in_s[ic * 64 + iy * 8 + ix] *
                 w_s[oc * 144 + ic * 9 + ky * 3 + kx];
        }
      }
    }
    sum += fmaxf(acc, 0.0f);
  }
  // reduce across the 8 threads sharing oc (aligned 8-lane groups in wave32)
  for (int off = 4; off >= 1; off >>= 1) sum += __shfl_xor(sum, off, 8);
  if (pg == 0) feats[img * 32 + oc] = sum * (1.0f / 64.0f);
}

// ------ GNN MLP + q/k rotation + attention temperature (alpha) -------------
__global__ __launch_bounds__(256) void gnn_kernel(
    const float* __restrict__ feats, const float* __restrict__ w1,
    const float* __restrict__ w2, const float* __restrict__ R,
    const float* __restrict__ ent, float* __restrict__ g,
    float* __restrict__ q, float* __restrict__ kk, float* __restrict__ alpha)
{
  __shared__ float w1_s[1024];   // (32,32)
  __shared__ float w2_s[512];    // (16,32)
  __shared__ float R_s[16];      // (4,4)
  const int t = threadIdx.x;
  const int row = blockIdx.x * 256 + t;
  for (int i = t; i < 1024; i += 256) w1_s[i] = w1[i];
  for (int i = t; i < 512; i += 256) w2_s[i] = w2[i];
  if (t < 16) R_s[t] = R[t];
  if (row == 0)
    alpha[0] = (cosf(ent[0]) + cosf(ent[1]) + cosf(ent[2])) * (1.0f / 3.0f) * 0.25f;
  __syncthreads();

  float f[32];
#pragma unroll
  for (int i = 0; i < 32; ++i) f[i] = feats[row * 32 + i];
  float a[32];
  for (int i = 0; i < 32; ++i) {
    float s = 0.0f;
#pragma unroll
    for (int j = 0; j < 32; ++j) s += f[j] * w1_s[i * 32 + j];
    a[i] = tanhf(s);
  }
  float gg[16];
  for (int i = 0; i < 16; ++i) {
    float s = 0.0f;
#pragma unroll
    for (int j = 0; j < 32; ++j) s += a[j] * w2_s[i * 32 + j];
    gg[i] = tanhf(s);
    g[row * 16 + i] = gg[i];
  }
  // q = (g.reshape(4,4) @ R).flatten ; k = (g.reshape(4,4) @ R^T).flatten
#pragma unroll
  for (int aa = 0; aa < 4; ++aa)
#pragma unroll
    for (int bb = 0; bb < 4; ++bb) {
      float sq = 0.0f, sk = 0.0f;
#pragma unroll
      for (int c = 0; c < 4; ++c) {
        sq += gg[aa * 4 + c] * R_s[c * 4 + bb];
        sk += gg[aa * 4 + c] * R_s[bb * 4 + c];
      }
      q[row * 16 + aa * 4 + bb] = sq;
      kk[row * 16 + aa * 4 + bb] = sk;
    }
}

// ------------ flash attention via V_WMMA_F32_16X16X4_F32, 2 phases ---------
// One wave32 owns a 16-row tile. Phase 1: sweep all 16-col chunks computing
// S = alpha * q k^T (4 WMMA/chunk) and a per-lane running max (lane (r,n)
// covers all columns == n mod 16, so the post-sweep 16-lane reduction gives
// the exact row max). Phase 2: re-sweep, P = exp(S - M) with fixed M (no
// rescaling), accumulate denominator as per-lane partials, O += P·g via
// 4 WMMA/chunk with P relaid C->A layout through per-wave LDS.
__global__ __launch_bounds__(128) void attn_kernel(
    const float* __restrict__ g, const float* __restrict__ q,
    const float* __restrict__ kp, const float* __restrict__ alpha_p,
    float* __restrict__ attn)
{
  __shared__ float lds[4 * 256];
  const int lane = threadIdx.x & 31;
  const int wave = threadIdx.x >> 5;
  const int tile = blockIdx.x * 4 + wave;
  const int m0 = tile * 16;
  float* P = &lds[wave * 256];
  const int half = lane >> 4;    // K sub-block selector for A/B operands
  const int n = lane & 15;       // col for B/C/D, row for A
  const float alpha = alpha_p[0];

  // A operand: q rows m0..m0+15 (stays in registers across both sweeps)
  v2f aq[4];
#pragma unroll
  for (int s = 0; s < 4; ++s) {
    const int kd = 4 * s + 2 * half;
    aq[s].x = q[(m0 + n) * 16 + kd];
    aq[s].y = q[(m0 + n) * 16 + kd + 1];
  }

  // ---- phase 1: per-lane running max of scaled logits (no shuffles) ----
  float pmax[8];
#pragma unroll
  for (int r = 0; r < 8; ++r) pmax[r] = -3.0e38f;

  for (int j0 = 0; j0 < BATCH; j0 += 16) {
    v2f bk[4];
#pragma unroll
    for (int s = 0; s < 4; ++s) {
      const int kd = 4 * s + 2 * half;
      bk[s].x = kp[(j0 + n) * 16 + kd];
      bk[s].y = kp[(j0 + n) * 16 + kd + 1];
    }
    v8f c = {};
#pragma unroll
    for (int s = 0; s < 4; ++s)
      c = __builtin_amdgcn_wmma_f32_16x16x4_f32(false, aq[s], false, bk[s],
                                                (short)0, c, false, false);
#pragma unroll
    for (int r = 0; r < 8; ++r) pmax[r] = fmaxf(pmax[r], alpha * c[r]);
  }
  // one exact row-max reduction per tile (row r+8*half across a 16-lane group)
#pragma unroll
  for (int r = 0; r < 8; ++r)
    for (int off = 8; off >= 1; off >>= 1)
      pmax[r] = fmaxf(pmax[r], __shfl_xor(pmax[r], off, 16));

  // ---- phase 2: P = exp(S - M), l += rowsum(P), O += P * g ----
  float lpart[8];
  v8f oacc = {};
#pragma unroll
  for (int r = 0; r < 8; ++r) lpart[r] = 0.0f;

  for (int j0 = 0; j0 < BATCH; j0 += 16) {
    v2f bk[4];
#pragma unroll
    for (int s = 0; s < 4; ++s) {
      const int kd = 4 * s + 2 * half;
      bk[s].x = kp[(j0 + n) * 16 + kd];
      bk[s].y = kp[(j0 + n) * 16 + kd + 1];
    }
    v8f c = {};
#pragma unroll
    for (int s = 0; s < 4; ++s)
      c = __builtin_amdgcn_wmma_f32_16x16x4_f32(false, aq[s], false, bk[s],
                                                (short)0, c, false, false);
#pragma unroll
    for (int r = 0; r < 8; ++r) {
      const float p = __expf(alpha * c[r] - pmax[r]);
      lpart[r] += p;
      P[(r + 8 * half) * 16 + n] = p;   // stash P tile in C/D layout
    }
    asm volatile("s_wait_dscnt 0" ::: "memory");

    // reload P in A layout; B operand is the g tile of this column chunk
    v2f ap[4], bg[4];
#pragma unroll
    for (int s = 0; s < 4; ++s) {
      const int kd = 4 * s + 2 * half;
      ap[s].x = P[n * 16 + kd];
      ap[s].y = P[n * 16 + kd + 1];
      bg[s].x = g[(j0 + kd) * 16 + n];
      bg[s].y = g[(j0 + kd + 1) * 16 + n];
    }
#pragma unroll
    for (int s = 0; s < 4; ++s)
      oacc = __builtin_amdgcn_wmma_f32_16x16x4_f32(false, ap[s], false, bg[s],
                                                   (short)0, oacc, false, false);
  }

  // one denominator reduction per tile, then normalize and write out
#pragma unroll
  for (int r = 0; r < 8; ++r) {
    for (int off = 8; off >= 1; off >>= 1)
      lpart[r] += __shfl_xor(lpart[r], off, 16);
    attn[(m0 + r + 8 * half) * 16 + n] = oacc[r] / lpart[r];
  }
}

// ------- reduce + (identity fraud layers) + classifier -> pre-BN out -------
__global__ __launch_bounds__(256) void post_kernel(
    const float* __restrict__ g, const float* __restrict__ attn,
    const float* __restrict__ rw, const float* __restrict__ rb,
    const float* __restrict__ cw, const float* __restrict__ cb,
    float* __restrict__ pre)
{
  const int row = blockIdx.x * 256 + threadIdx.x;
  float r0 = rb[0], r1 = rb[1];
#pragma unroll
  for (int i = 0; i < 16; ++i) {
    const float gv = g[row * 16 + i];
    const float av = attn[row * 16 + i];
    r0 += gv * rw[i] + av * rw[16 + i];
    r1 += gv * rw[32 + i] + av * rw[48 + i];
  }
  // fraud layers with all-zero params are exact identity
  pre[row * 2 + 0] = cw[0] * r0 + cw[1] * r1 + cb[0];
  pre[row * 2 + 1] = cw[2] * r0 + cw[3] * r1 + cb[1];
}

// --------------------- batch-norm (batch stats, biased var) ----------------
__global__ __launch_bounds__(256) void bn_kernel(
    const float* __restrict__ pre, const float* __restrict__ gamma,
    const float* __restrict__ beta, float* __restrict__ out)
{
  __shared__ float s0[256], s1[256], q0[256], q1[256];
  __shared__ float mu0s, mu1s, sc0s, sc1s;
  const int t = threadIdx.x;
  float a0 = 0, a1 = 0, b0 = 0, b1 = 0;
  for (int r = t; r < BATCH; r += 256) {
    const float x0 = pre[2 * r], x1 = pre[2 * r + 1];
    a0 += x0; a1 += x1; b0 += x0 * x0; b1 += x1 * x1;
  }
  s0[t] = a0; s1[t] = a1; q0[t] = b0; q1[t] = b1;
  __syncthreads();
  for (int off = 128; off > 0; off >>= 1) {
    if (t < off) {
      s0[t] += s0[t + off]; s1[t] += s1[t + off];
      q0[t] += q0[t + off]; q1[t] += q1[t + off];
    }
    __syncthreads();
  }
  if (t == 0) {
    const float m0 = s0[0] / (float)BATCH, m1 = s1[0] / (float)BATCH;
    const float v0 = q0[0] / (float)BATCH - m0 * m0;
    const float v1 = q1[0] / (float)BATCH - m1 * m1;
    mu0s = m0; mu1s = m1;
    sc0s = rsqrtf(v0 + 1e-5f) * gamma[0];
    sc1s = rsqrtf(v1 + 1e-5f) * gamma[1];
  }
  __syncthreads();
  const float m0 = mu0s, m1 = mu1s, c0 = sc0s, c1 = sc1s;
  const float be0 = beta[0], be1 = beta[1];
  for (int r = t; r < BATCH; r += 256) {
    out[2 * r]     = (pre[2 * r]     - m0) * c0 + be0;
    out[2 * r + 1] = (pre[2 * r + 1] - m1) * c1 + be1;
  }
}

// ---------------------------------------------------------------------------
extern "C" void kernel_launch(void* const* d_in, const int* in_sizes, int n_in,
                              void* d_out, int out_size, void* d_ws, size_t ws_size,
                              hipStream_t stream) {
  const float* x      = (const float*)d_in[0];
  const float* c1w    = (const float*)d_in[1];
  const float* c1b    = (const float*)d_in[2];
  const float* c2w    = (const float*)d_in[3];
  const float* c2b    = (const float*)d_in[4];
  const float* c3w    = (const float*)d_in[5];
  const float* c3b    = (const float*)d_in[6];
  const float* w1     = (const float*)d_in[7];
  const float* w2     = (const float*)d_in[8];
  const float* rot    = (const float*)d_in[9];
  const float* ent    = (const float*)d_in[10];
  const float* rw     = (const float*)d_in[11];
  const float* rb     = (const float*)d_in[12];
  const float* cw     = (const float*)d_in[13];
  const float* cb     = (const float*)d_in[14];
  const float* gamma  = (const float*)d_in[15];
  const float* beta   = (const float*)d_in[16];
  float* out = (float*)d_out;

  float* ws    = (float*)d_ws;       // requires ~52 MB of scratch
  float* h1    = ws + WS_H1;
  float* h2    = ws + WS_H2;
  float* feats = ws + WS_FEATS;
  float* g     = ws + WS_G;
  float* q     = ws + WS_Q;
  float* k     = ws + WS_K;
  float* attn  = ws + WS_ATTN;
  float* pre   = ws + WS_PRE;
  float* alpha = ws + WS_ALPHA;

  conv1_kernel<<<BATCH, 256, 0, stream>>>(x, c1w, c1b, h1);
  conv2_kernel<<<BATCH, 256, 0, stream>>>(h1, c2w, c2b, h2);
  conv3_kernel<<<BATCH, 256, 0, stream>>>(h2, c3w, c3b, feats);
  gnn_kernel<<<BATCH / 256, 256, 0, stream>>>(feats, w1, w2, rot, ent, g, q, k, alpha);
  attn_kernel<<<BATCH / 64, 128, 0, stream>>>(g, q, k, alpha, attn);   // 256 tiles / 4 waves
  post_kernel<<<BATCH / 256, 256, 0, stream>>>(g, attn, rw, rb, cw, cb, pre);
  bn_kernel<<<1, 256, 0, stream>>>(pre, gamma, beta, out);
}